// VADetector_68135361183900
// MI455X (gfx1250) — compile-verified
//
#include <hip/hip_runtime.h>
#include <stdint.h>

// ---------------------------------------------------------------------------
// 16-state Viterbi detector for MI455X (gfx1250, wave32).
//
//  * lane = batch row; 16 path metrics kept in 16 VGPRs per lane
//  * T split into NCHUNK chunks with WARM warm-up steps (path-merge trick)
//    -> 64 row-groups x 16 chunks = 1024 waves
//  * y tiles (32 rows x 32 t) staged to LDS via global_load_async_to_lds_b128
//    (CDNA5 async path, ASYNCcnt / s_wait_asynccnt), double buffered
//  * the reference's  symbols @ h.T  (16x4 * 4x1) is computed with one
//    V_WMMA_F32_16X16X4_F32 per wave
//  * decoded bits packed in a per-lane u32, redistributed with v_readlane,
//    stored as coalesced 128B rows
// ---------------------------------------------------------------------------

#define N_STATES 16
#define T_LEN 4096
#define B_ROWS 2048
#define TILE 32
#define NCHUNK 16
#define CHUNK (T_LEN / NCHUNK) /* 256 */
#define WARM 96                /* warm-up steps, multiple of TILE */
#define WAVES_PER_BLOCK 8
#define YSTRIDE 36 /* floats per LDS row: 144B (16B aligned, low bank conflict) */

typedef __attribute__((ext_vector_type(2))) float v2f;
typedef __attribute__((ext_vector_type(8))) float v8f;

__device__ __forceinline__ void async_load_tile(const float* __restrict__ y,
                                                int r0, int tbase,
                                                uint32_t lds_base_bytes, int lane) {
  const int sub = lane >> 3;        // row-in-quad 0..3
  const int col = (lane & 7) << 2;  // float col 0,4,...,28
#pragma unroll
  for (int i = 0; i < 8; ++i) {
    const int row = i * 4 + sub;
    uint64_t ga = (uint64_t)(uintptr_t)(y + (size_t)(r0 + row) * T_LEN + tbase + col);
    uint32_t la = lds_base_bytes + (uint32_t)(row * YSTRIDE + col) * 4u;
    // CDNA5 async copy: global -> LDS, tracked by ASYNCcnt (no VGPR data).
    asm volatile("global_load_async_to_lds_b128 %0, %1, off"
                 :: "v"(la), "v"(ga) : "memory");
  }
}

__global__ void __launch_bounds__(256)
va_detector_kernel(const float* __restrict__ y,
                   const float* __restrict__ h,
                   float* __restrict__ out) {
  __shared__ __align__(16) float ybuf[WAVES_PER_BLOCK][2][32 * YSTRIDE];

  const int lane = threadIdx.x & 31;
  const int w = threadIdx.x >> 5;
  const int wave_id = blockIdx.x * WAVES_PER_BLOCK + w;
  const int rg = wave_id / NCHUNK;  // row group (32 rows)
  const int ck = wave_id % NCHUNK;  // time chunk
  const int r0 = rg * 32;
  const int t_emit = ck * CHUNK;
  const int t_begin = (ck == 0) ? 0 : (t_emit - WARM);
  const int t_end = t_emit + CHUNK;
  const int ntiles = (t_end - t_begin) / TILE;

  const float h0 = h[0], h1 = h[1], h2 = h[2], h3 = h[3];

  // ---- state priors sp[s] = symbols(s,:) . h   (reference: symbols @ h.T) ----
  float sp[N_STATES];
#if __has_builtin(__builtin_amdgcn_wmma_f32_16x16x4_f32)
  {
    // A (16x4 f32): VGPR0 = K0 (lanes0-15) / K2 (lanes16-31); VGPR1 = K1/K3.
    const int m = lane & 15;
    const int kh = (lane >> 4) << 1;  // 0 or 2
    v2f a, b;
    a.x = ((m >> (3 - kh)) & 1) ? -1.0f : 1.0f;
    a.y = ((m >> (3 - (kh + 1))) & 1) ? -1.0f : 1.0f;
    b.x = (kh == 0) ? h0 : h2;  // B row K broadcast over all N
    b.y = (kh == 0) ? h1 : h3;
    v8f c = {};
    v8f d = __builtin_amdgcn_wmma_f32_16x16x4_f32(
        /*neg_a=*/false, a, /*neg_b=*/false, b,
        /*c_mod=*/(short)0, c, /*reuse_a=*/false, /*reuse_b=*/false);
    // D[v]: lanes0-15 -> M=v, lanes16-31 -> M=v+8 (all N columns equal).
    const bool hi = lane >= 16;
#pragma unroll
    for (int v = 0; v < 8; ++v) {
      float mine = d[v];
      float other = __shfl_xor(mine, 16, 32);
      sp[v] = hi ? other : mine;
      sp[v + 8] = hi ? mine : other;
    }
  }
#else
#pragma unroll
  for (int s = 0; s < N_STATES; ++s)
    sp[s] = ((s & 8) ? -h0 : h0) + ((s & 4) ? -h1 : h1) +
            ((s & 2) ? -h2 : h2) + ((s & 1) ? -h3 : h3);
#endif

  // Effective per-step prior: sp^2/2 - y*sp  (uniform y^2/2 - C term dropped:
  // it shifts all states equally, so argmin / min selection are unchanged).
  float q[N_STATES];
#pragma unroll
  for (int s = 0; s < N_STATES; ++s) q[s] = 0.5f * sp[s] * sp[s];

  float p[N_STATES];
#pragma unroll
  for (int s = 0; s < N_STATES; ++s) p[s] = 0.0f;

  const uint32_t lds0 = (uint32_t)(uintptr_t)(&ybuf[w][0][0]);
  const uint32_t lds1 = (uint32_t)(uintptr_t)(&ybuf[w][1][0]);

  async_load_tile(y, r0, t_begin, lds0, lane);

  for (int k = 0; k < ntiles; ++k) {
    const int tbase = t_begin + k * TILE;
    const bool more = (k + 1) < ntiles;
    if (more)  // prefetch next tile into the other buffer
      async_load_tile(y, r0, tbase + TILE, (k & 1) ? lds0 : lds1, lane);
    if (more)
      asm volatile("s_wait_asynccnt 0x8" ::: "memory");  // current tile done
    else
      asm volatile("s_wait_asynccnt 0x0" ::: "memory");

    const float* yrow = &ybuf[w][k & 1][0] + lane * YSTRIDE;  // transposed read
    uint32_t mask = 0;
#pragma unroll 4
    for (int tt = 0; tt < TILE; ++tt) {
      const float yv = yrow[tt];  // ds_load_b32, per-lane time sample

      // decoded bit: parity of argmin(in_prob) == (min over odd < min over even)
      float me = fminf(fminf(fminf(p[0], p[2]), fminf(p[4], p[6])),
                       fminf(fminf(p[8], p[10]), fminf(p[12], p[14])));
      float mo = fminf(fminf(fminf(p[1], p[3]), fminf(p[5], p[7])),
                       fminf(fminf(p[9], p[11]), fminf(p[13], p[15])));
      mask |= ((mo < me) ? 1u : 0u) << tt;

      float mm[N_STATES];
#pragma unroll
      for (int s = 0; s < N_STATES; ++s)
        mm[s] = fmaf(-yv, sp[s], p[s] + q[s]);
      // transition_table[s] = (2s % 16, (2s+1) % 16)
#pragma unroll
      for (int s = 0; s < N_STATES; ++s)
        p[s] = fminf(mm[(2 * s) & 15], mm[((2 * s) & 15) + 1]);
    }

    if (tbase >= t_emit) {  // skip warm-up region
#pragma unroll
      for (int i = 0; i < 32; ++i) {
        uint32_t sm = (uint32_t)__builtin_amdgcn_readlane((int)mask, i);
        out[(size_t)(r0 + i) * T_LEN + tbase + lane] = (float)((sm >> lane) & 1u);
      }
    }
  }
}

extern "C" void kernel_launch(void* const* d_in, const int* in_sizes, int n_in,
                              void* d_out, int out_size, void* d_ws, size_t ws_size,
                              hipStream_t stream) {
  (void)in_sizes; (void)n_in; (void)d_ws; (void)ws_size; (void)out_size;
  const float* y = (const float*)d_in[0];
  const float* h = (const float*)d_in[1];
  // d_in[2] (transition_table) is a fixed de Bruijn shift: tt[s] = (2s%16, (2s+1)%16),
  // hardcoded in the kernel.
  float* out = (float*)d_out;

  const int total_waves = (B_ROWS / 32) * NCHUNK;            // 1024
  const int blocks = total_waves / WAVES_PER_BLOCK;          // 128
  hipLaunchKernelGGL(va_detector_kernel, dim3(blocks), dim3(256), 0, stream,
                     y, h, out);
}